// EuclideanCodebook_62440234549775
// MI455X (gfx1250) — compile-verified
//
#include <hip/hip_runtime.h>

// ---------------------------------------------------------------------------
// EuclideanCodebook nearest-neighbor quantization for MI455X (gfx1250).
//   score[m][n] = ||e_n||^2 - 2 * <x_m, e_n>      (x^2 row-constant, dropped)
// GEMM term via V_WMMA_F32_16X16X4_F32 (exact f32 on the matrix pipe).
// Each wave32 owns a 16-row x block in A-layout VGPRs and loops the whole
// codebook from LDS-staged chunks. B operands + ||e||^2 are double-buffered
// one tile ahead so LDS/global latency overlaps the WMMA chain.
// ---------------------------------------------------------------------------

typedef float v2f __attribute__((ext_vector_type(2)));
typedef float v4f __attribute__((ext_vector_type(4)));
typedef float v8f __attribute__((ext_vector_type(8)));

#define N_TOTAL      65536
#define K_CB         4096
#define D_DIM        64
#define WAVES_PER_WG 8
#define WG_THREADS   256
#define CHUNK_ROWS   128         // embed rows staged per LDS chunk
#define TILES_PER_CHUNK (CHUNK_ROWS / 16)
#define LDS_STRIDE   68          // 64 + 4 pad floats -> conflict-free banks

// ----------------------------- ||e||^2 kernel ------------------------------
__global__ __launch_bounds__(256)
void esq_kernel(const float* __restrict__ embed, float* __restrict__ esq) {
    int k = blockIdx.x * blockDim.x + threadIdx.x;
    if (k < K_CB) {
        const float* p = embed + (size_t)k * D_DIM;
        float s = 0.f;
#pragma unroll
        for (int d = 0; d < D_DIM; ++d) s += p[d] * p[d];
        esq[k] = s;
    }
}

// ------------------------------ main kernel --------------------------------
__global__ __launch_bounds__(WG_THREADS)
void vq_wmma_kernel(const float* __restrict__ x,
                    const float* __restrict__ embed,
                    const float* __restrict__ esq,
                    float* __restrict__ out) {
    __shared__ float lds[CHUNK_ROWS * LDS_STRIDE];   // 34816 B

    const int tid    = threadIdx.x;
    const int wave   = tid >> 5;       // 0..7
    const int lane   = tid & 31;
    const int lane16 = lane & 15;
    const int hi     = lane >> 4;      // 0 or 1 (lane half)

    // 16 x-rows per wave, 128 per workgroup.
    const int rowBase = (blockIdx.x * WAVES_PER_WG + wave) * 16;

    // ---- load x block into WMMA A-layout registers (reused for all K) ----
    // A (16x4 f32): VGPR0 = K0 (lanes 0-15) / K0+2 (lanes 16-31); VGPR1 = +1/+3.
    v2f xa[16];
    const float* xrow = x + (size_t)(rowBase + lane16) * D_DIM;
#pragma unroll
    for (int s = 0; s < 16; ++s) {
        const float* p = xrow + 4 * s + 2 * hi;
        xa[s][0] = p[0];
        xa[s][1] = p[1];
    }

    // Running argmin state in C-layout: VGPR r, lanes 0-15 -> row r,
    // lanes 16-31 -> row r+8; lane%16 is the code slot within a 16-wide tile.
    v8f best_d;
    int best_i[8];
#pragma unroll
    for (int r = 0; r < 8; ++r) { best_d[r] = 3.4e38f; best_i[r] = 0; }

    // B-tile fetch into registers (tile-local, fully unrolled -> no scratch).
    auto load_btile = [&](int t, v2f (&b)[16]) {
        const float* brow = &lds[(t * 16 + lane16) * LDS_STRIDE + 2 * hi];
#pragma unroll
        for (int s = 0; s < 16; ++s) {
            b[s][0] = brow[4 * s];
            b[s][1] = brow[4 * s + 1];
        }
    };

    for (int chunk = 0; chunk < K_CB; chunk += CHUNK_ROWS) {
        __syncthreads();
        // ---- coalesced stage of CHUNK_ROWS x 64 f32 into padded LDS ----
#pragma unroll
        for (int i = 0; i < CHUNK_ROWS / 16; ++i) {
            int r = i * 16 + (tid >> 4);
            int c = (tid & 15) * 4;
            v4f v = *(const v4f*)(embed + (size_t)(chunk + r) * D_DIM + c);
            *(v4f*)(&lds[r * LDS_STRIDE + c]) = v;
        }
        __syncthreads();

        // Pull the next chunk (32 KB, L2-resident) toward the WGP while the
        // WMMA chain below runs: 256 threads x 128 B covers it.
        if (chunk + CHUNK_ROWS < K_CB) {
            const char* np =
                (const char*)(embed + (size_t)(chunk + CHUNK_ROWS) * D_DIM) +
                (size_t)tid * 128;
            __builtin_prefetch(np, 0, 1);
        }

        // ---- software-pipelined tiles: load t+1 while t's WMMAs run ----
        v2f b0[16], b1[16];
        float es0, es1;
        load_btile(0, b0);
        es0 = esq[chunk + lane16];
        es1 = 0.f;

#pragma unroll
        for (int t = 0; t < TILES_PER_CHUNK; ++t) {
            v2f (&bc)[16] = (t & 1) ? b1 : b0;   // current tile operands
            v2f (&bn)[16] = (t & 1) ? b0 : b1;   // next tile buffer
            const float esc = (t & 1) ? es1 : es0;

            if (t + 1 < TILES_PER_CHUNK) {
                load_btile(t + 1, bn);
                float en = esq[chunk + (t + 1) * 16 + lane16];
                if (t & 1) es0 = en; else es1 = en;
            }

            v8f acc = {};
#pragma unroll
            for (int s = 0; s < 16; ++s) {
                acc = __builtin_amdgcn_wmma_f32_16x16x4_f32(
                    /*neg_a=*/false, xa[s], /*neg_b=*/false, bc[s],
                    /*c_mod=*/(short)0, acc, /*reuse_a=*/false, /*reuse_b=*/false);
            }

            const int nidx = chunk + t * 16 + lane16;
#pragma unroll
            for (int r = 0; r < 8; ++r) {
                float score = esc - 2.0f * acc[r];
                if (score < best_d[r]) { best_d[r] = score; best_i[r] = nidx; }
            }
        }
    }

    // ---- row-wise argmin across the 16 code slots (stay inside half) ----
#pragma unroll
    for (int r = 0; r < 8; ++r) {
        float d = best_d[r];
        int   i = best_i[r];
#pragma unroll
        for (int m = 8; m >= 1; m >>= 1) {
            float od = __shfl_xor(d, m, 32);
            int   oi = __shfl_xor(i, m, 32);
            if (od < d || (od == d && oi < i)) { d = od; i = oi; }
        }
        best_i[r] = i;   // all 16 lanes of the half now agree
    }

    // ---- outputs: quantize gather [N,64] then indices [N] (as f32) ----
    float* qout = out;
    float* iout = out + (size_t)N_TOTAL * D_DIM;
#pragma unroll
    for (int r = 0; r < 8; ++r) {
        int row = rowBase + r + 8 * hi;
        int idx = best_i[r];
        v4f v = *(const v4f*)(embed + (size_t)idx * D_DIM + lane16 * 4);
        *(v4f*)(qout + (size_t)row * D_DIM + lane16 * 4) = v;
        if (lane16 == 0) iout[row] = (float)idx;
    }
}

// ------------------------------ launcher -----------------------------------
extern "C" void kernel_launch(void* const* d_in, const int* in_sizes, int n_in,
                              void* d_out, int out_size, void* d_ws, size_t ws_size,
                              hipStream_t stream) {
    const float* x     = (const float*)d_in[0];   // [N, D] f32
    const float* embed = (const float*)d_in[1];   // [1, K, D] f32
    float*       out   = (float*)d_out;           // [N*D] quantize + [N] indices
    float*       esq   = (float*)d_ws;            // K floats scratch

    hipLaunchKernelGGL(esq_kernel, dim3(K_CB / 256), dim3(256), 0, stream,
                       embed, esq);
    hipLaunchKernelGGL(vq_wmma_kernel, dim3(N_TOTAL / (16 * WAVES_PER_WG)),
                       dim3(WG_THREADS), 0, stream, x, embed, esq, out);
}